// HDBVLUT_13477607375182
// MI455X (gfx1250) — compile-verified
//
#include <hip/hip_runtime.h>
#include <stdint.h>

// HDBVLUT 4x super-resolution, fused 16-pass (4 kernels x 4 rotations).
// Single output write per element; image tile staged to LDS with CDNA5
// async global->LDS copies (ASYNCcnt); LUT rows gathered from L2.

#define LVL 17
#define IMH 256
#define IMW 256
#define OUTW (IMW * 4)
#define TILE_W 32
#define TILE_H 8
#define HALO_W (TILE_W + 4)              // 36
#define HALO_H (TILE_H + 4)              // 12
#define HALO_N (HALO_W * HALO_H)         // 432

__global__ __launch_bounds__(TILE_W * TILE_H)
void hdbv_lut4x_kernel(const int* __restrict__ img,
                       const float* __restrict__ hW,
                       const float* __restrict__ dW,
                       const float* __restrict__ bW,
                       const float* __restrict__ vW,
                       float* __restrict__ out)
{
    __shared__ int s_img[HALO_N];

    const int plane = blockIdx.z;                 // b*C + c, 24 planes
    const int ty0   = blockIdx.y * TILE_H;
    const int tx0   = blockIdx.x * TILE_W;
    const int* imgP = img + (size_t)plane * (IMH * IMW);

    // ---- stage haloed tile (edge-clamped) into LDS with async copies ----
    const int tid = threadIdx.y * TILE_W + threadIdx.x;
    for (int cell = tid; cell < HALO_N; cell += TILE_W * TILE_H) {
        int ly = cell / HALO_W;
        int lx = cell - ly * HALO_W;
        int gy = ty0 + ly - 2; gy = gy < 0 ? 0 : (gy > IMH - 1 ? IMH - 1 : gy);
        int gx = tx0 + lx - 2; gx = gx < 0 ? 0 : (gx > IMW - 1 ? IMW - 1 : gx);
        unsigned ldsOff = (unsigned)(uintptr_t)(&s_img[cell]);         // low 32b of flat = LDS offset
        unsigned long long gaddr = (unsigned long long)(uintptr_t)(imgP + gy * IMW + gx);
        asm volatile("global_load_async_to_lds_b32 %0, %1, off"
                     :: "v"(ldsOff), "v"(gaddr) : "memory");
    }
    asm volatile("s_wait_asynccnt 0" ::: "memory");
    __syncthreads();

    // ---- 5x5 clamped window in registers ----
    int win[5][5];
#pragma unroll
    for (int i = 0; i < 5; ++i)
#pragma unroll
        for (int j = 0; j < 5; ++j)
            win[i][j] = s_img[(threadIdx.y + i) * HALO_W + (threadIdx.x + j)];

    const int a = win[2][2];                      // offset (0,0) for every pass
    float acc[16];
#pragma unroll
    for (int e = 0; e < 16; ++e) acc[e] = 0.0f;

    // {bdy,bdx,cdy,cdx} for ktypes h, d, b, v
    const int offs[4][4] = { {0,1,0,2}, {1,1,2,2}, {1,2,2,1}, {1,0,2,0} };

#pragma unroll
    for (int r = 0; r < 4; ++r) {
#pragma unroll
        for (int k = 0; k < 4; ++k) {
            const float* wk = (k == 0) ? hW : (k == 1) ? dW : (k == 2) ? bW : vW;
            int bdy = offs[k][0], bdx = offs[k][1];
            int cdy = offs[k][2], cdx = offs[k][3];
            // rotated-frame (dy,dx) -> original-frame window delta
            int br, bc, cr, cc;
            if (r == 0)      { br =  bdy; bc =  bdx; cr =  cdy; cc =  cdx; }
            else if (r == 1) { br =  bdx; bc = -bdy; cr =  cdx; cc = -cdy; }
            else if (r == 2) { br = -bdy; bc = -bdx; cr = -cdy; cc = -cdx; }
            else             { br = -bdx; bc =  bdy; cr = -cdx; cc =  cdy; }
            int bv = win[2 + br][2 + bc];
            int cv = win[2 + cr][2 + cc];
            int idx = (a * LVL + bv) * LVL + cv;

            const float4* wp = (const float4*)(wk + (size_t)idx * 16);  // 64B row, 16B aligned
            float4 q0 = wp[0], q1 = wp[1], q2 = wp[2], q3 = wp[3];
            float wr[16] = { q0.x,q0.y,q0.z,q0.w, q1.x,q1.y,q1.z,q1.w,
                             q2.x,q2.y,q2.z,q2.w, q3.x,q3.y,q3.z,q3.w };

            // accumulate with intra-block back-rotation permutation
#pragma unroll
            for (int e = 0; e < 16; ++e) {
                int u = e >> 2, v = e & 3, s;
                if (r == 0)      s = u * 4 + v;
                else if (r == 1) s = (3 - v) * 4 + u;
                else if (r == 2) s = (3 - u) * 4 + (3 - v);
                else             s = v * 4 + (3 - u);
                acc[e] += wr[s];
            }
        }
    }

    // ---- single coalesced write of the 4x4 block, scaled by 1/4 ----
    const int h = ty0 + threadIdx.y;
    const int w = tx0 + threadIdx.x;
    float* outP = out + (size_t)plane * (size_t)(IMH * 4) * OUTW;
#pragma unroll
    for (int u = 0; u < 4; ++u) {
        float4 o;
        o.x = acc[u * 4 + 0] * 0.25f;
        o.y = acc[u * 4 + 1] * 0.25f;
        o.z = acc[u * 4 + 2] * 0.25f;
        o.w = acc[u * 4 + 3] * 0.25f;
        *(float4*)(outP + (size_t)(4 * h + u) * OUTW + 4 * w) = o;
    }
}

extern "C" void kernel_launch(void* const* d_in, const int* in_sizes, int n_in,
                              void* d_out, int out_size, void* d_ws, size_t ws_size,
                              hipStream_t stream) {
    const int*   img = (const int*)d_in[0];
    const float* hW  = (const float*)d_in[1];
    const float* dW  = (const float*)d_in[2];
    const float* bW  = (const float*)d_in[3];
    const float* vW  = (const float*)d_in[4];
    float* out = (float*)d_out;

    dim3 block(TILE_W, TILE_H, 1);
    dim3 grid(IMW / TILE_W, IMH / TILE_H, 8 * 3);
    hdbv_lut4x_kernel<<<grid, block, 0, stream>>>(img, hW, dW, bW, vW, out);
}